// WeightedMSELoss_8306466751246
// MI455X (gfx1250) — compile-verified
//
#include <hip/hip_runtime.h>
#include <math.h>

// WeightedMSELoss-with-Y-norm: loss = mean_{b,x,z} || mask*(net-gt) ||_2 over Y
// Shapes: [8,1,128,128,128] f32. Pure HBM-bound streaming reduction:
// 201.3 MB in / 4 B out -> ~8.6 us floor at 23.3 TB/s.

typedef __attribute__((ext_vector_type(2))) float v2f;
typedef __attribute__((ext_vector_type(4))) float v4f;
typedef __attribute__((ext_vector_type(8))) float v8f;

#define NB 8
#define NX 128
#define NY 128
#define NZ 128
#define INV_COUNT (1.0f / (float)(NB * NX * NZ))

// Sum a value across all 32 lanes of a wave using V_WMMA_F32_16X16X4_F32.
// A = [lane vals | zeros] (16x4), B = ones (4x16)  ->  D[m,n] = rowsum_A(m).
// Sum of D column n over all 16 rows = sum of ALL A elements = wave sum,
// independent of A's internal VGPR striping. Per the documented C/D layout,
// lane n (n<16) holds rows 0..7 in its 8 D VGPRs and lane n+16 holds rows
// 8..15, so one xor-16 shuffle completes the column sum.
// Requires EXEC == all 1s (call from a fully-active wave).
__device__ __forceinline__ float wave32_sum_wmma(float x) {
  v2f a;
  a[0] = x;
  a[1] = 0.0f;
  v2f b;
  b[0] = 1.0f;
  b[1] = 1.0f;
  v8f c = {};
  c = __builtin_amdgcn_wmma_f32_16x16x4_f32(
      /*neg_a=*/false, a, /*neg_b=*/false, b,
      /*c_mod=*/(short)0, c, /*reuse_a=*/false, /*reuse_b=*/false);
  float t = c[0] + c[1] + c[2] + c[3] + c[4] + c[5] + c[6] + c[7];
  return t + __shfl_xor(t, 16, 32);
}

__global__ __launch_bounds__(128) void wmse_ynorm_kernel(
    const float* __restrict__ gt, const float* __restrict__ net,
    const float* __restrict__ msk, float* __restrict__ out) {
  __shared__ v4f red[128];

  const int t = threadIdx.x;      // 0..127
  const int lane = t & 31;        // z group: lane owns z = 4*lane .. 4*lane+3
  const int wv = t >> 5;          // y chunk: wave owns y = 32*wv .. 32*wv+31

  // blockIdx.x indexes (b, x) slabs: element offset b*X*Y*Z + x*Y*Z
  const size_t slab = (size_t)blockIdx.x * (size_t)(NY * NZ);
  const size_t base = slab + (size_t)wv * 32u * NZ + (size_t)(lane << 2);

  const float* gp = gt + base;
  const float* np = net + base;
  const float* mp = msk + base;

  // Per-lane partial sum over this wave's 32 y-rows; 4 independent z columns.
  v4f acc = (v4f)0.0f;
#pragma unroll 8
  for (int y = 0; y < 32; ++y) {
    const size_t o = (size_t)y * NZ;
    v4f g = __builtin_nontemporal_load((const v4f*)(gp + o));
    v4f n = __builtin_nontemporal_load((const v4f*)(np + o));
    v4f m = __builtin_nontemporal_load((const v4f*)(mp + o));
    v4f d = m * (n - g);
    acc += d * d;
  }

  red[t] = acc;
  __syncthreads();

  if (wv == 0) {  // wave 0 fully active -> EXEC all 1s for WMMA
    // Complete the Y reduction across the 4 waves, then sqrt per z.
    v4f s = red[lane] + red[lane + 32] + red[lane + 64] + red[lane + 96];
    float lsum = sqrtf(s[0]) + sqrtf(s[1]) + sqrtf(s[2]) + sqrtf(s[3]);
    // Sum the 32 lane values (128 z norms) via WMMA reduction.
    float wsum = wave32_sum_wmma(lsum);
    if (lane == 0) atomicAdd(out, wsum * INV_COUNT);
  }
}

extern "C" void kernel_launch(void* const* d_in, const int* in_sizes, int n_in,
                              void* d_out, int out_size, void* d_ws, size_t ws_size,
                              hipStream_t stream) {
  const float* gt = (const float*)d_in[0];   // ground_truth
  const float* net = (const float*)d_in[1];  // net_output
  const float* msk = (const float*)d_in[2];  // mask
  float* out = (float*)d_out;                // scalar f32

  // d_out is poisoned; zero it inside the captured stream before atomics.
  hipMemsetAsync(out, 0, sizeof(float), stream);

  wmse_ynorm_kernel<<<NB * NX, 128, 0, stream>>>(gt, net, msk, out);
}